// CPCLossNetwork_35854386987767
// MI455X (gfx1250) — compile-verified
//
#include <hip/hip_runtime.h>
#include <hip/hip_bf16.h>

typedef __attribute__((ext_vector_type(2))) float v2f;
typedef __attribute__((ext_vector_type(8))) float v8f;
typedef __attribute__((ext_vector_type(4))) int   v4i;

typedef __attribute__((address_space(1))) v4i* gv4_t;  // global src for async-LDS
typedef __attribute__((address_space(3))) v4i* lv4_t;  // LDS dst for async-LDS

#define N_   8
#define T_   512
#define C_   256
#define K_   12
#define M_   128
#define TP_  500
#define KC_  (K_ * C_)      // 3072
#define ROWS_ (N_ * TP_)    // 4000

// ---------------------------------------------------------------------------
// Kernel 1: Az = Z @ W^T   (4000 x 3072) = (4000 x 256) x (256 x 3072)
// 8 waves per block; each wave produces one 16x16 tile via v_wmma_f32_16x16x4_f32.
// grid: 250 m-tiles * (192/8) n-tile-groups = 6000 blocks.
// ---------------------------------------------------------------------------
__global__ __launch_bounds__(256) void az_gemm_kernel(const float* __restrict__ latent,
                                                      const float* __restrict__ W,
                                                      float* __restrict__ az) {
  const int wave  = threadIdx.x >> 5;
  const int lane  = threadIdx.x & 31;
  const int mtile = blockIdx.x / 24;                 // 0..249
  const int ntile = (blockIdx.x % 24) * 8 + wave;    // 0..191
  const int half  = lane >> 4;                       // 0 or 1
  const int l16   = lane & 15;

  // A-matrix row: latent row of flattened Z (rows may cross n-boundaries)
  const int r  = mtile * 16 + l16;                   // 0..3999
  const int nb = r / TP_;
  const int tb = r % TP_;
  const float* arow = latent + (size_t)(nb * T_ + tb) * C_;

  // B-matrix column j (== row j of W, since B = W^T)
  const int j = ntile * 16 + l16;
  const float* brow = W + (size_t)j * C_;

  v8f acc = {};
  for (int kk = 0; kk < C_; kk += 4) {
    const int kb = kk + 2 * half;                    // per-lane K base
    v2f a = *(const v2f*)(arow + kb);                // A[l16, kb..kb+1]
    v2f b = *(const v2f*)(brow + kb);                // B[kb..kb+1, l16]
    acc = __builtin_amdgcn_wmma_f32_16x16x4_f32(false, a, false, b,
                                                (short)0, acc, false, false);
  }

  // D layout: vgpr v, lane l -> row (v + 8*half), col l16
  float* out = az + (size_t)(mtile * 16 + half * 8) * KC_ + (size_t)ntile * 16 + l16;
#pragma unroll
  for (int v = 0; v < 8; ++v)
    out[(size_t)v * KC_] = acc[v];
}

// ---------------------------------------------------------------------------
// Kernel 2: per (n,t): scores = Az_nt(16x256,padded) x Phi^T(256x128),
// then logsumexp over negatives, positive dots, softplus, block-sum -> atomic.
// One block (8 waves) per (n,t); wave w handles negatives [16w, 16w+16).
// Az(n,t) is staged into LDS with GLOBAL_LOAD_ASYNC_TO_LDS_B128 (ASYNCcnt).
// ---------------------------------------------------------------------------
__global__ __launch_bounds__(256) void score_loss_kernel(const float* __restrict__ latent,
                                                         const int* __restrict__ samps,
                                                         const float* __restrict__ az,
                                                         float* __restrict__ accum) {
  __shared__ float azs[16 * C_];      // 16 KB: Az(n,t), rows 12..15 zero-padded
  __shared__ float sc[16 * M_];       //  8 KB: score matrix (16 x 128)
  __shared__ float pmax[K_ * 16];     // partial max   (k x 16)
  __shared__ float psum[K_ * 16];     // partial sumexp(k x 16)
  __shared__ float pdot[K_ * 16];     // partial dot   (k x 16)
  __shared__ float red[16];

  const int b    = blockIdx.x;        // 0..3999 == n*Tp + t
  const int n    = b / TP_;
  const int t    = b % TP_;
  const int tid  = threadIdx.x;
  const int wave = tid >> 5;
  const int lane = tid & 31;
  const int half = lane >> 4;
  const int l16  = lane & 15;

  // ---- async-stage Az(n,t): 12x256 f32 = 12 KB = 768 x b128, 3 per thread ----
  const float* azsrc = az + (size_t)b * KC_;
#pragma unroll
  for (int i = 0; i < 3; ++i) {
    const int idx = tid + 256 * i;            // b128 chunk index, 0..767
    __builtin_amdgcn_global_load_async_to_lds_b128(
        (gv4_t)(azsrc + idx * 4), (lv4_t)(azs + idx * 4), 0, 0);
  }
  // zero-pad rows 12..15 with plain ds stores (DScnt, drained by barrier)
  for (int i = K_ * C_ + tid; i < 16 * C_; i += 256)
    azs[i] = 0.0f;
  asm volatile("s_wait_asynccnt 0x0" ::: "memory");
  __syncthreads();

  // ---- this lane's negative row (gather through L2; latent is only 4 MB) ----
  const int m    = wave * 16 + l16;                       // 0..127
  const int srow = samps[(size_t)b * M_ + m];             // row into (N*T, C)
  const float* prow = latent + (size_t)srow * C_;

  v8f acc = {};
  for (int kk = 0; kk < C_; kk += 4) {
    const int kb = kk + 2 * half;                         // even -> 8B aligned
    v2f a  = *(const v2f*)(azs + l16 * C_ + kb);          // ds_load_b64
    v2f bb = *(const v2f*)(prow + kb);                    // global float2
    acc = __builtin_amdgcn_wmma_f32_16x16x4_f32(false, a, false, bb,
                                                (short)0, acc, false, false);
  }

  // scatter this wave's 16x16 tile into the 16x128 score matrix
#pragma unroll
  for (int v = 0; v < 8; ++v)
    sc[(v + 8 * half) * M_ + wave * 16 + l16] = acc[v];
  __syncthreads();

  // ---- parallel reductions: 12 k-rows x 16 partial-threads each ----
  if (tid < K_ * 16) {
    const int k = tid >> 4;
    const int j = tid & 15;

    // partial logsumexp over 8 negatives: m = 8j .. 8j+7
    float mx = -3.402823466e38f;
#pragma unroll
    for (int u = 0; u < 8; ++u) mx = fmaxf(mx, sc[k * M_ + j * 8 + u]);
    float s = 0.0f;
#pragma unroll
    for (int u = 0; u < 8; ++u) s += expf(sc[k * M_ + j * 8 + u] - mx);
    pmax[tid] = mx;
    psum[tid] = s;

    // partial positive dot over 16 channels: c = 16j .. 16j+15
    const float* pk = latent + (size_t)(n * T_ + (1 + t + k)) * C_ + j * 16;
    float d = 0.0f;
#pragma unroll
    for (int u = 0; u < 16; ++u) d += azs[k * C_ + j * 16 + u] * pk[u];
    pdot[tid] = d;
  }
  __syncthreads();

  // ---- combine per k: lse, num, softplus ----
  if (tid < K_) {
    const int k = tid;
    float gmax = -3.402823466e38f;
    for (int j = 0; j < 16; ++j) gmax = fmaxf(gmax, pmax[k * 16 + j]);
    float s = 0.0f;
    for (int j = 0; j < 16; ++j) s += psum[k * 16 + j] * expf(pmax[k * 16 + j] - gmax);
    const float lse = gmax + logf(s);

    float num = 0.0f;
    for (int j = 0; j < 16; ++j) num += pdot[k * 16 + j];

    const float d = lse - num;
    // stable softplus: logaddexp(num, lse) - num
    red[k] = fmaxf(d, 0.0f) + log1pf(expf(-fabsf(d)));
  }
  __syncthreads();

  if (tid == 0) {
    float s = 0.0f;
    for (int k = 0; k < K_; ++k) s += red[k];
    atomicAdd(accum, s);   // GLOBAL_ATOMIC_ADD_F32
  }
}

__global__ void init_kernel(float* accum) { *accum = 0.0f; }

__global__ void final_kernel(const float* __restrict__ accum, float* __restrict__ out) {
  *out = *accum / (float)(ROWS_ * K_);
}

// ---------------------------------------------------------------------------
extern "C" void kernel_launch(void* const* d_in, const int* in_sizes, int n_in,
                              void* d_out, int out_size, void* d_ws, size_t ws_size,
                              hipStream_t stream) {
  const float* latent = (const float*)d_in[0];   // (8, 512, 256) f32
  const float* W      = (const float*)d_in[1];   // (3072, 256)   f32
  const int*   samps  = (const int*)d_in[2];     // (8*500*128)   i32

  float* az    = (float*)d_ws;                       // 4000*3072 f32 = 49.15 MB
  float* accum = az + (size_t)ROWS_ * KC_;           // 1 f32 accumulator

  init_kernel<<<1, 1, 0, stream>>>(accum);
  az_gemm_kernel<<<250 * 24, 256, 0, stream>>>(latent, W, az);
  score_loss_kernel<<<ROWS_, 256, 0, stream>>>(latent, samps, az, accum);
  final_kernel<<<1, 1, 0, stream>>>(accum, (float*)d_out);
}